// BaseEventWarping_9174050144274
// MI455X (gfx1250) — compile-verified
//
#include <hip/hip_runtime.h>
#include <math.h>
#include <stdint.h>

// Problem constants (match reference)
#define H_ 480
#define W_ 640
#define B_ 4
#define P_ 10
#define N_ 400000
#define HW_ (H_ * W_)
#define TILE_H 8

__device__ __forceinline__ float charb6(float a) { return sqrtf(a * a + 1e-6f); }

__device__ __forceinline__ float waveSum(float v) {
#pragma unroll
  for (int o = 16; o > 0; o >>= 1) v += __shfl_down(v, o, 32);
  return v;
}

// block-wide reduce of two values -> atomicAdd into g2[0..1]
__device__ __forceinline__ void block_reduce2(float v0, float v1, float* g2) {
  __shared__ float r2[2];
  if (threadIdx.x < 2) r2[threadIdx.x] = 0.f;
  __syncthreads();
  v0 = waveSum(v0);
  v1 = waveSum(v1);
  if ((threadIdx.x & 31) == 0) {
    atomicAdd(&r2[0], v0);
    atomicAdd(&r2[1], v1);
  }
  __syncthreads();
  if (threadIdx.x < 2) atomicAdd(&g2[threadIdx.x], r2[threadIdx.x]);
}

// ---------------------------------------------------------------------------
// Kernel 1: bilinear scatter of events into weight image + polarity ts images
// ---------------------------------------------------------------------------
__global__ __launch_bounds__(256) void k_scatter(const float* __restrict__ we,
                                                 const float* __restrict__ pol,
                                                 const float* __restrict__ ts,
                                                 float* __restrict__ accw,
                                                 float* __restrict__ acctp,
                                                 float* __restrict__ acctn) {
  const int i = blockIdx.x * blockDim.x + threadIdx.x;
  const int b = blockIdx.y;
  if (i >= N_) return;

  const float2 yx = ((const float2*)we)[(size_t)b * N_ + i];
  const float y = yx.x, x = yx.y;
  const float p = ((const float2*)pol)[(size_t)b * 4 * N_ + i].x;  // pos in {0,1}
  const float t = 1.0f - fabsf(1.0f - ts[(size_t)b * 4 * N_ + i]); // norm_ts

  const float ty = floorf(y), lx = floorf(x);
  const float fy = y - ty, fx = x - lx;
  const int iy0 = (int)ty, ix0 = (int)lx;
  const float wy[2] = {1.f - fy, fy};
  const float wx[2] = {1.f - fx, fx};

  float* aw = accw + (size_t)b * HW_;
  float* at = (p > 0.5f ? acctp : acctn) + (size_t)b * HW_;

#pragma unroll
  for (int dy = 0; dy < 2; dy++) {
    const int iy = iy0 + dy;
    const bool iby = (iy >= 0) && (iy <= H_ - 1);
    const int cy = min(max(iy, 0), H_ - 1);
#pragma unroll
    for (int dx = 0; dx < 2; dx++) {
      const int ix = ix0 + dx;
      const bool ibx = (ix >= 0) && (ix <= W_ - 1);
      float w = wy[dy] * wx[dx];
      if (!(iby && ibx)) w = 0.f;
      if (w != 0.f) {
        const int cx = min(max(ix, 0), W_ - 1);
        const int idx = cy * W_ + cx;
        atomicAdd(&aw[idx], w);
        atomicAdd(&at[idx], w * t);
      }
    }
  }
}

// ---------------------------------------------------------------------------
// Kernel 2: focus-loss per-batch reduction: sum(tp^2+tn^2), count(w != 0)
// ---------------------------------------------------------------------------
__global__ __launch_bounds__(256) void k_focus(const float* __restrict__ accw,
                                               const float* __restrict__ acctp,
                                               const float* __restrict__ acctn,
                                               float* __restrict__ fsc) {
  const int i = blockIdx.x * blockDim.x + threadIdx.x;
  const int b = blockIdx.y;
  float sq = 0.f, nz = 0.f;
  if (i < HW_) {
    const size_t o = (size_t)b * HW_ + i;
    const float tp = acctp[o], tn = acctn[o];
    sq = tp * tp + tn * tn;
    nz = (accw[o] != 0.f) ? 1.f : 0.f;
  }
  block_reduce2(sq, nz, &fsc[b * 2]);
}

// ---------------------------------------------------------------------------
// Kernel 3: temporal smoothing: warp grid by flow[j], gather flow[j+1]
// ---------------------------------------------------------------------------
__global__ __launch_bounds__(256) void k_temporal(const float* __restrict__ fxg,
                                                  const float* __restrict__ fyg,
                                                  float* __restrict__ tsc) {
  const int i = blockIdx.x * blockDim.x + threadIdx.x;
  const int j = blockIdx.y;  // 0..P-2
  const int b = blockIdx.z;
  float num = 0.f, den = 0.f;
  if (i < HW_) {
    const size_t pl0 = ((size_t)(b * P_ + j)) * HW_;
    const size_t pl1 = pl0 + HW_;
    const float fx0 = fxg[pl0 + i];
    const float fy0 = fyg[pl0 + i];
    const int y = i / W_;
    const int x = i - y * W_;
    const float ly = (float)y + fy0;
    const float lx = (float)x + fx0;
    if (ly >= 0.f && ly <= (float)(H_ - 1) && lx >= 0.f && lx <= (float)(W_ - 1)) {
      const float ty = floorf(ly), tx = floorf(lx);
      const float fy = ly - ty, fx = lx - tx;
      const int iy0 = (int)ty, ix0 = (int)tx;
      const int iy1 = min(iy0 + 1, H_ - 1);
      const int ix1 = min(ix0 + 1, W_ - 1);
      const float w00 = (1.f - fy) * (1.f - fx);
      const float w01 = (1.f - fy) * fx;
      const float w10 = fy * (1.f - fx);
      const float w11 = fy * fx;
      const float* Fx = fxg + pl1;
      const float* Fy = fyg + pl1;
      const int i00 = iy0 * W_ + ix0, i01 = iy0 * W_ + ix1;
      const int i10 = iy1 * W_ + ix0, i11 = iy1 * W_ + ix1;
      const float sx = Fx[i00] * w00 + Fx[i01] * w01 + Fx[i10] * w10 + Fx[i11] * w11;
      const float sy = Fy[i00] * w00 + Fy[i01] * w01 + Fy[i10] * w10 + Fy[i11] * w11;
      const float dx = fx0 - sx, dy = fy0 - sy;
      num = sqrtf(dy * dy + 1e-9f) + sqrtf(dx * dx + 1e-9f);
      den = 1.f;
    }
  }
  block_reduce2(num, den, &tsc[(size_t)(b * (P_ - 1) + j) * 2]);
}

// ---------------------------------------------------------------------------
// Kernel 4: spatial smoothing, async global->LDS tile staging (CDNA5 path).
// Each block stages (TILE_H+1) rows x 640 of flow_x & flow_y into LDS via
// global_load_async_to_lds_b128 (ASYNCcnt-tracked, bypasses VGPRs), waits with
// s_wait_asynccnt, then computes the 4-direction Charbonnier stencil from LDS.
// ---------------------------------------------------------------------------
__global__ __launch_bounds__(256) void k_spatial(const float* __restrict__ fxg,
                                                 const float* __restrict__ fyg,
                                                 float* __restrict__ ssc) {
  __shared__ float sfx[(TILE_H + 1) * W_];
  __shared__ float sfy[(TILE_H + 1) * W_];
  __shared__ float red[4];
  const int tid = threadIdx.x;
  const int r0 = blockIdx.x * TILE_H;
  const int p = blockIdx.y;
  const int b = blockIdx.z;
  const int nrows = (r0 + TILE_H < H_) ? (TILE_H + 1) : (H_ - r0);  // +halo row
  const size_t plane = ((size_t)(b * P_ + p)) * HW_ + (size_t)r0 * W_;
  const float* gx = fxg + plane;
  const float* gy = fyg + plane;

  const int nvec = nrows * (W_ / 4);  // 16B vectors per array
  for (int i = tid; i < nvec; i += 256) {
    uint32_t lax = (uint32_t)(uintptr_t)(&sfx[i * 4]);
    uint64_t gax = (uint64_t)(uintptr_t)(gx + i * 4);
    asm volatile("global_load_async_to_lds_b128 %0, %1, off" ::"v"(lax), "v"(gax)
                 : "memory");
    uint32_t lay = (uint32_t)(uintptr_t)(&sfy[i * 4]);
    uint64_t gay = (uint64_t)(uintptr_t)(gy + i * 4);
    asm volatile("global_load_async_to_lds_b128 %0, %1, off" ::"v"(lay), "v"(gay)
                 : "memory");
  }
  if (tid < 4) red[tid] = 0.f;
#if __has_builtin(__builtin_amdgcn_s_wait_asynccnt)
  __builtin_amdgcn_s_wait_asynccnt(0);
#else
  asm volatile("s_wait_asynccnt 0" ::: "memory");
#endif
  __syncthreads();

  float sdx = 0.f, sdy = 0.f, sdd = 0.f, sdu = 0.f;
  for (int i = tid; i < TILE_H * W_; i += 256) {
    const int lyr = i / W_;
    const int x = i - lyr * W_;
    const int y = r0 + lyr;
    const float a = sfx[lyr * W_ + x];
    const float c = sfy[lyr * W_ + x];
    const bool hx = (x < W_ - 1);
    const bool hy = (y < H_ - 1);
    if (hx) sdx += charb6(a - sfx[lyr * W_ + x + 1]) + charb6(c - sfy[lyr * W_ + x + 1]);
    if (hy) sdy += charb6(a - sfx[(lyr + 1) * W_ + x]) + charb6(c - sfy[(lyr + 1) * W_ + x]);
    if (hx && hy) {
      sdd += charb6(a - sfx[(lyr + 1) * W_ + x + 1]) + charb6(c - sfy[(lyr + 1) * W_ + x + 1]);
      sdu += charb6(sfx[(lyr + 1) * W_ + x] - sfx[lyr * W_ + x + 1]) +
             charb6(sfy[(lyr + 1) * W_ + x] - sfy[lyr * W_ + x + 1]);
    }
  }
  sdx = waveSum(sdx);
  sdy = waveSum(sdy);
  sdd = waveSum(sdd);
  sdu = waveSum(sdu);
  if ((tid & 31) == 0) {
    atomicAdd(&red[0], sdx);
    atomicAdd(&red[1], sdy);
    atomicAdd(&red[2], sdd);
    atomicAdd(&red[3], sdu);
  }
  __syncthreads();
  if (tid < 4) atomicAdd(&ssc[(size_t)(b * P_ + p) * 4 + tid], red[tid]);
}

// ---------------------------------------------------------------------------
// Kernel 5: combine all partial scalars into the final loss
// ---------------------------------------------------------------------------
__global__ void k_final(const float* __restrict__ fsc, const float* __restrict__ tsc,
                        const float* __restrict__ ssc, float* __restrict__ out) {
  if (threadIdx.x != 0 || blockIdx.x != 0) return;
  float focus = 0.f;
  for (int b = 0; b < B_; b++) focus += fsc[b * 2 + 0] / (fsc[b * 2 + 1] + 1e-9f);

  float temporal = 0.f;
  for (int b = 0; b < B_; b++) {
    float acc = 0.f;
    for (int j = 0; j < P_ - 1; j++) {
      const float* t = &tsc[(b * (P_ - 1) + j) * 2];
      acc += t[0] / (t[1] + 1e-9f);
    }
    temporal += acc / (float)(P_ - 1);
  }

  const float inv_dx = 1.f / ((float)H_ * (float)(W_ - 1));
  const float inv_dy = 1.f / ((float)(H_ - 1) * (float)W_);
  const float inv_dd = 1.f / ((float)(H_ - 1) * (float)(W_ - 1));
  float spatial = 0.f;
  for (int b = 0; b < B_; b++) {
    float acc = 0.f;
    for (int p = 0; p < P_; p++) {
      const float* s = &ssc[(b * P_ + p) * 4];
      acc += (s[0] * inv_dx + s[1] * inv_dy + s[2] * inv_dd + s[3] * inv_dd) * 0.25f;
    }
    spatial += acc / (float)P_;
  }
  out[0] = focus + temporal + spatial;
}

// ---------------------------------------------------------------------------
extern "C" void kernel_launch(void* const* d_in, const int* in_sizes, int n_in,
                              void* d_out, int out_size, void* d_ws, size_t ws_size,
                              hipStream_t stream) {
  (void)in_sizes; (void)n_in; (void)out_size; (void)ws_size;
  const float* we  = (const float*)d_in[0];  // [B,N,2]  (y,x)
  const float* pol = (const float*)d_in[1];  // [B,4N,2]
  const float* ts  = (const float*)d_in[2];  // [B,4N,1]
  const float* fx  = (const float*)d_in[3];  // [B,P,H,W]
  const float* fy  = (const float*)d_in[4];  // [B,P,H,W]

  float* ws = (float*)d_ws;
  float* accw  = ws;                        // [B,HW]
  float* acctp = ws + (size_t)B_ * HW_;     // [B,HW]
  float* acctn = ws + 2 * (size_t)B_ * HW_; // [B,HW]
  float* fsc   = ws + 3 * (size_t)B_ * HW_; // [B][sq,cnt]
  float* tsc   = fsc + B_ * 2;              // [B*(P-1)][num,den]
  float* ssc   = tsc + B_ * (P_ - 1) * 2;   // [B*P][dx,dy,ddr,dur]

  const size_t zbytes =
      (3 * (size_t)B_ * HW_ + B_ * 2 + B_ * (P_ - 1) * 2 + B_ * P_ * 4) * sizeof(float);
  hipMemsetAsync(d_ws, 0, zbytes, stream);

  const dim3 blk(256);
  k_scatter<<<dim3((N_ + 255) / 256, B_), blk, 0, stream>>>(we, pol, ts, accw, acctp, acctn);
  k_focus<<<dim3((HW_ + 255) / 256, B_), blk, 0, stream>>>(accw, acctp, acctn, fsc);
  k_temporal<<<dim3((HW_ + 255) / 256, P_ - 1, B_), blk, 0, stream>>>(fx, fy, tsc);
  k_spatial<<<dim3(H_ / TILE_H, P_, B_), blk, 0, stream>>>(fx, fy, ssc);
  k_final<<<1, 32, 0, stream>>>(fsc, tsc, ssc, (float*)d_out);
}